// LabelPropagation_8950711846049
// MI455X (gfx1250) — compile-verified
//
#include <hip/hip_runtime.h>
#include <math.h>

// Problem constants (from reference setup_inputs)
#define NTOT   1100          // 100 support + 1000 query
#define NPAD   1152          // padded: 72 tiles of 16 = 36 pairs = 18 quad-strips
#define DMODEL 640
#define HWSP   100           // 10*10 spatial
#define TOPK_K 10
#define EPSF   2.220446049250313e-16f   // np.finfo(float64).eps
#define RED_BLOCKS 512

typedef __attribute__((ext_vector_type(2))) float v2f;
typedef __attribute__((ext_vector_type(8))) float v8f;

#define WMMA_F32(a, b, c) \
  __builtin_amdgcn_wmma_f32_16x16x4_f32(false, (a), false, (b), (short)0, (c), false, false)

// ---------------------------------------------------------------------------
// 1) spatial mean into padded x [NPAD, DMODEL]; padded rows written as zero
// ---------------------------------------------------------------------------
__global__ void mean_kernel(const float* __restrict__ sup,
                            const float* __restrict__ qry,
                            float* __restrict__ x) {
  int wave = (int)((blockIdx.x * blockDim.x + threadIdx.x) >> 5);
  int lane = threadIdx.x & 31;
  if (wave >= NPAD * DMODEL) return;          // wave-uniform
  int n = wave / DMODEL, c = wave - n * DMODEL;
  if (n >= NTOT) {                            // zero padding rows (wave-uniform)
    if (lane == 0) x[(size_t)n * DMODEL + c] = 0.f;
    return;
  }
  const float* src = (n < 100)
      ? (sup + ((size_t)n * DMODEL + c) * HWSP)
      : (qry + ((size_t)(n - 100) * DMODEL + c) * HWSP);
  float s = 0.f;
  for (int i = lane; i < HWSP; i += 32) s += src[i];
  for (int off = 16; off > 0; off >>= 1) s += __shfl_down(s, off, 32);
  if (lane == 0) x[(size_t)n * DMODEL + c] = s * (1.0f / HWSP);
}

// ---------------------------------------------------------------------------
// 2) WMMA fp32 GEMM, C = A * B^T, 32x64 output per wave (2x4 tiles),
//    software-pipelined: next K-step's operands load while current WMMAs run.
// ---------------------------------------------------------------------------
__global__ void wmma_gemm_nt8(float* __restrict__ C,
                              const float* __restrict__ A,
                              const float* __restrict__ B,
                              int tilesM2, int tilesN4, int K,
                              int lda, int ldb, int ldc, int Mstore) {
  int lane  = threadIdx.x & 31;
  int strip = (int)(blockIdx.x * (blockDim.x >> 5) + (threadIdx.x >> 5));
  if (strip >= tilesM2 * tilesN4) return;     // wave-uniform: EXEC all-1 below
  int tm = strip / tilesN4, ts = strip - tm * tilesN4;
  int kh = (lane >> 4) << 1;                  // lane-half K offset: 0 or 2
  const float* A0 = A + (size_t)(tm * 32 + (lane & 15)) * lda + kh;
  const float* A1 = A0 + (size_t)16 * lda;
  const float* B0 = B + (size_t)(ts * 64 + (lane & 15)) * ldb + kh;
  const float* B1 = B0 + (size_t)16 * ldb;
  const float* B2 = B0 + (size_t)32 * ldb;
  const float* B3 = B0 + (size_t)48 * ldb;
  v8f c00 = {}, c01 = {}, c02 = {}, c03 = {};
  v8f c10 = {}, c11 = {}, c12 = {}, c13 = {};
  v2f a0 = *(const v2f*)A0, a1 = *(const v2f*)A1;
  v2f b0 = *(const v2f*)B0, b1 = *(const v2f*)B1;
  v2f b2 = *(const v2f*)B2, b3 = *(const v2f*)B3;
  for (int k0 = 4; k0 < K; k0 += 4) {
    v2f na0 = *(const v2f*)(A0 + k0), na1 = *(const v2f*)(A1 + k0);
    v2f nb0 = *(const v2f*)(B0 + k0), nb1 = *(const v2f*)(B1 + k0);
    v2f nb2 = *(const v2f*)(B2 + k0), nb3 = *(const v2f*)(B3 + k0);
    c00 = WMMA_F32(a0, b0, c00); c01 = WMMA_F32(a0, b1, c01);
    c02 = WMMA_F32(a0, b2, c02); c03 = WMMA_F32(a0, b3, c03);
    c10 = WMMA_F32(a1, b0, c10); c11 = WMMA_F32(a1, b1, c11);
    c12 = WMMA_F32(a1, b2, c12); c13 = WMMA_F32(a1, b3, c13);
    a0 = na0; a1 = na1; b0 = nb0; b1 = nb1; b2 = nb2; b3 = nb3;
  }
  c00 = WMMA_F32(a0, b0, c00); c01 = WMMA_F32(a0, b1, c01);
  c02 = WMMA_F32(a0, b2, c02); c03 = WMMA_F32(a0, b3, c03);
  c10 = WMMA_F32(a1, b0, c10); c11 = WMMA_F32(a1, b1, c11);
  c12 = WMMA_F32(a1, b2, c12); c13 = WMMA_F32(a1, b3, c13);
  int cn = ts * 64 + (lane & 15);
  int mb = tm * 32 + ((lane >> 4) << 3);
  #pragma unroll
  for (int r = 0; r < 8; ++r) {
    int cm0 = mb + r, cm1 = mb + 16 + r;
    if (cm0 < Mstore) {
      float* crow = C + (size_t)cm0 * ldc + cn;
      crow[0] = c00[r]; crow[16] = c01[r]; crow[32] = c02[r]; crow[48] = c03[r];
    }
    if (cm1 < Mstore) {
      float* crow = C + (size_t)cm1 * ldc + cn;
      crow[0] = c10[r]; crow[16] = c11[r]; crow[32] = c12[r]; crow[48] = c13[r];
    }
  }
}

// ---------------------------------------------------------------------------
// 3) WMMA fp32 GEMM, C = A * B, 32x64 output per wave (2x4 tiles), pipelined
// ---------------------------------------------------------------------------
__global__ void wmma_gemm_nn8(float* __restrict__ C,
                              const float* __restrict__ A,
                              const float* __restrict__ B,
                              int tilesM2, int tilesN4, int K,
                              int lda, int ldb, int ldc, int Mstore) {
  int lane  = threadIdx.x & 31;
  int strip = (int)(blockIdx.x * (blockDim.x >> 5) + (threadIdx.x >> 5));
  if (strip >= tilesM2 * tilesN4) return;
  int tm = strip / tilesN4, ts = strip - tm * tilesN4;
  int kh = (lane >> 4) << 1;
  const float* A0 = A + (size_t)(tm * 32 + (lane & 15)) * lda + kh;
  const float* A1 = A0 + (size_t)16 * lda;
  const float* Bc = B + (size_t)(ts * 64 + (lane & 15)) + (size_t)kh * ldb;
  v8f c00 = {}, c01 = {}, c02 = {}, c03 = {};
  v8f c10 = {}, c11 = {}, c12 = {}, c13 = {};
  v2f a0 = *(const v2f*)A0, a1 = *(const v2f*)A1;
  v2f b0, b1, b2, b3;
  {
    const float* lo = Bc;                     // row kh
    const float* hi = Bc + ldb;               // row kh+1
    b0.x = lo[0];  b0.y = hi[0];  b1.x = lo[16]; b1.y = hi[16];
    b2.x = lo[32]; b2.y = hi[32]; b3.x = lo[48]; b3.y = hi[48];
  }
  for (int k0 = 4; k0 < K; k0 += 4) {
    v2f na0 = *(const v2f*)(A0 + k0), na1 = *(const v2f*)(A1 + k0);
    const float* lo = Bc + (size_t)k0 * ldb;
    const float* hi = lo + ldb;
    v2f nb0, nb1, nb2, nb3;
    nb0.x = lo[0];  nb0.y = hi[0];  nb1.x = lo[16]; nb1.y = hi[16];
    nb2.x = lo[32]; nb2.y = hi[32]; nb3.x = lo[48]; nb3.y = hi[48];
    c00 = WMMA_F32(a0, b0, c00); c01 = WMMA_F32(a0, b1, c01);
    c02 = WMMA_F32(a0, b2, c02); c03 = WMMA_F32(a0, b3, c03);
    c10 = WMMA_F32(a1, b0, c10); c11 = WMMA_F32(a1, b1, c11);
    c12 = WMMA_F32(a1, b2, c12); c13 = WMMA_F32(a1, b3, c13);
    a0 = na0; a1 = na1; b0 = nb0; b1 = nb1; b2 = nb2; b3 = nb3;
  }
  c00 = WMMA_F32(a0, b0, c00); c01 = WMMA_F32(a0, b1, c01);
  c02 = WMMA_F32(a0, b2, c02); c03 = WMMA_F32(a0, b3, c03);
  c10 = WMMA_F32(a1, b0, c10); c11 = WMMA_F32(a1, b1, c11);
  c12 = WMMA_F32(a1, b2, c12); c13 = WMMA_F32(a1, b3, c13);
  int cn = ts * 64 + (lane & 15);
  int mb = tm * 32 + ((lane >> 4) << 3);
  #pragma unroll
  for (int r = 0; r < 8; ++r) {
    int cm0 = mb + r, cm1 = mb + 16 + r;
    if (cm0 < Mstore) {
      float* crow = C + (size_t)cm0 * ldc + cn;
      crow[0] = c00[r]; crow[16] = c01[r]; crow[32] = c02[r]; crow[48] = c03[r];
    }
    if (cm1 < Mstore) {
      float* crow = C + (size_t)cm1 * ldc + cn;
      crow[0] = c10[r]; crow[16] = c11[r]; crow[32] = c12[r]; crow[48] = c13[r];
    }
  }
}

// ---------------------------------------------------------------------------
// 4) row squared-norms of emb_q (rows < NTOT): one wave per row
// ---------------------------------------------------------------------------
__global__ void rowsq_kernel(const float* __restrict__ q, float* __restrict__ sq) {
  int wave = (int)((blockIdx.x * blockDim.x + threadIdx.x) >> 5);
  int lane = threadIdx.x & 31;
  if (wave >= NTOT) return;
  const float* r = q + (size_t)wave * DMODEL;
  float s = 0.f;
  for (int k = lane; k < DMODEL; k += 32) { float t = r[k]; s += t * t; }
  for (int off = 16; off > 0; off >>= 1) s += __shfl_down(s, off, 32);
  if (lane == 0) sq[wave] = s;
}

// ---------------------------------------------------------------------------
// 5) w[i,j] = max(sq[i]+sq[j]-2*G[i,j], 0); diag and padded region = 0
// ---------------------------------------------------------------------------
__global__ void build_w_kernel(float* __restrict__ w, const float* __restrict__ sq) {
  int idx = (int)(blockIdx.x * blockDim.x + threadIdx.x);
  if (idx >= NPAD * NPAD) return;
  int i = idx / NPAD, j = idx - i * NPAD;
  float val = 0.f;
  if (i < NTOT && j < NTOT && i != j)
    val = fmaxf(sq[i] + sq[j] - 2.f * w[idx], 0.f);
  w[idx] = val;
}

// ---------------------------------------------------------------------------
// 6) deterministic two-stage reductions for mean / variance (ddof=1, off-diag)
// ---------------------------------------------------------------------------
__global__ void red_sum_kernel(const float* __restrict__ w, float* __restrict__ part) {
  __shared__ float sm[256];
  float s = 0.f;
  for (int idx = (int)(blockIdx.x * 256 + threadIdx.x); idx < NPAD * NPAD;
       idx += RED_BLOCKS * 256) s += w[idx];   // padded entries & diag are 0
  sm[threadIdx.x] = s; __syncthreads();
  for (int off = 128; off > 0; off >>= 1) {
    if ((int)threadIdx.x < off) sm[threadIdx.x] += sm[threadIdx.x + off];
    __syncthreads();
  }
  if (threadIdx.x == 0) part[blockIdx.x] = sm[0];
}

__global__ void fin_mean_kernel(const float* __restrict__ part, float* __restrict__ stats) {
  if (threadIdx.x || blockIdx.x) return;
  float s = 0.f;
  for (int b = 0; b < RED_BLOCKS; ++b) s += part[b];      // fixed order
  stats[0] = s / (float)(NTOT * NTOT - NTOT);
}

__global__ void red_var_kernel(const float* __restrict__ w,
                               const float* __restrict__ stats,
                               float* __restrict__ part) {
  __shared__ float sm[256];
  float mean = stats[0];
  float s = 0.f;
  for (int idx = (int)(blockIdx.x * 256 + threadIdx.x); idx < NPAD * NPAD;
       idx += RED_BLOCKS * 256) {
    int i = idx / NPAD, j = idx - i * NPAD;
    if (i < NTOT && j < NTOT && i != j) { float d = w[idx] - mean; s += d * d; }
  }
  sm[threadIdx.x] = s; __syncthreads();
  for (int off = 128; off > 0; off >>= 1) {
    if ((int)threadIdx.x < off) sm[threadIdx.x] += sm[threadIdx.x + off];
    __syncthreads();
  }
  if (threadIdx.x == 0) part[blockIdx.x] = sm[0];
}

__global__ void fin_var_kernel(const float* __restrict__ part, float* __restrict__ stats) {
  if (threadIdx.x || blockIdx.x) return;
  float s = 0.f;
  for (int b = 0; b < RED_BLOCKS; ++b) s += part[b];
  float var = s / (float)(NTOT * NTOT - NTOT - 1);        // ddof = 1
  stats[1] = 0.5f / sqrtf(var);                           // exp(-w/(2*sigma))
}

// 7) w = exp(-w / (2*sqrt(var))) inside [NTOT,NTOT]; padded region forced to 0
__global__ void gauss_kernel(float* __restrict__ w, const float* __restrict__ stats) {
  int idx = (int)(blockIdx.x * blockDim.x + threadIdx.x);
  if (idx >= NPAD * NPAD) return;
  int i = idx / NPAD, j = idx - i * NPAD;
  w[idx] = (i < NTOT && j < NTOT) ? expf(-w[idx] * stats[1]) : 0.f;
}

// 8) mask bookkeeping (byte mask with NPAD row stride, re-zeroed every call)
__global__ void zero_mask_kernel(unsigned int* __restrict__ m, int n_words) {
  int idx = (int)(blockIdx.x * blockDim.x + threadIdx.x);
  if (idx < n_words) m[idx] = 0u;
}

__global__ void topk_mask_kernel(const float* __restrict__ w,
                                 unsigned char* __restrict__ mask) {
  int i = (int)(blockIdx.x * blockDim.x + threadIdx.x);
  if (i >= NTOT) return;
  float v[TOPK_K]; int id[TOPK_K];
  #pragma unroll
  for (int t = 0; t < TOPK_K; ++t) { v[t] = -3.4e38f; id[t] = 0; }
  const float* row = w + (size_t)i * NPAD;
  for (int j = 0; j < NTOT; ++j) {
    float x = row[j];
    int am = 0; float mn = v[0];
    #pragma unroll
    for (int t = 1; t < TOPK_K; ++t) if (v[t] < mn) { mn = v[t]; am = t; }
    if (x > mn) { v[am] = x; id[am] = j; }
  }
  for (int t = 0; t < TOPK_K; ++t) {
    int j = id[t];
    mask[(size_t)i * NPAD + j] = 1;   // same-value concurrent writes: benign
    mask[(size_t)j * NPAD + i] = 1;   // symmetric (mask + mask^T > 0)
  }
}

__global__ void apply_mask_kernel(float* __restrict__ w,
                                  const unsigned char* __restrict__ mask) {
  int idx = (int)(blockIdx.x * blockDim.x + threadIdx.x);
  if (idx >= NPAD * NPAD) return;
  w[idx] *= (float)mask[idx];
}

// 9) degree (masked w is symmetric: row-sum == column-sum), then D^{-1/2}
__global__ void deg_kernel(const float* __restrict__ w, float* __restrict__ deg) {
  int wave = (int)((blockIdx.x * blockDim.x + threadIdx.x) >> 5);
  int lane = threadIdx.x & 31;
  if (wave >= NTOT) return;
  const float* r = w + (size_t)wave * NPAD;
  float s = 0.f;
  for (int j = lane; j < NTOT; j += 32) s += r[j];
  for (int off = 16; off > 0; off >>= 1) s += __shfl_down(s, off, 32);
  if (lane == 0) deg[wave] = s;
}

__global__ void dsi_kernel(const float* __restrict__ deg, float* __restrict__ dsi) {
  int i = (int)(blockIdx.x * blockDim.x + threadIdx.x);
  if (i >= NTOT) return;
  dsi[i] = sqrtf(1.0f / (deg[i] + EPSF));
}

// 10) S = dsi_i * w_ij * dsi_j (guarded: padded region stays exactly 0)
__global__ void scale_w_kernel(float* __restrict__ w, const float* __restrict__ dsi) {
  int idx = (int)(blockIdx.x * blockDim.x + threadIdx.x);
  if (idx >= NPAD * NPAD) return;
  int i = idx / NPAD, j = idx - i * NPAD;
  if (i < NTOT && j < NTOT) w[idx] *= dsi[i] * dsi[j];
}

// ---------------------------------------------------------------------------
static inline size_t alignUp(size_t x) { return (x + 255) & ~(size_t)255; }

extern "C" void kernel_launch(void* const* d_in, const int* in_sizes, int n_in,
                              void* d_out, int out_size, void* d_ws, size_t ws_size,
                              hipStream_t stream) {
  const float* support = (const float*)d_in[0];   // [100,640,10,10]
  const float* query   = (const float*)d_in[2];   // [1000,640,10,10]
  const float* Wq      = (const float*)d_in[4];   // [640,640]
  const float* Wv      = (const float*)d_in[5];   // [640,640]
  float* out = (float*)d_out;                     // [1100,640]

  // workspace carve (~15.5 MB, fully rewritten every call; fits in L2)
  char* p = (char*)d_ws;
  float* x    = (float*)p; p += alignUp((size_t)NPAD * DMODEL * 4);
  float* eq   = (float*)p; p += alignUp((size_t)NPAD * DMODEL * 4);
  float* ev   = (float*)p; p += alignUp((size_t)NPAD * DMODEL * 4);
  float* w    = (float*)p; p += alignUp((size_t)NPAD * NPAD * 4);
  float* sq   = (float*)p; p += alignUp((size_t)NTOT * 4);
  float* deg  = (float*)p; p += alignUp((size_t)NTOT * 4);
  float* dsi  = (float*)p; p += alignUp((size_t)NTOT * 4);
  float* part = (float*)p; p += alignUp((size_t)RED_BLOCKS * 4);
  float* stats= (float*)p; p += alignUp(16);
  unsigned char* mask = (unsigned char*)p;        // NPAD*NPAD bytes

  dim3 blk(256);
  const int WPB  = 256 / 32;                      // waves per block
  const int NNP  = NPAD * NPAD;
  const int TM2  = NPAD / 32;                     // 36 row tile-pairs
  const int TN_D = DMODEL / 64;                   // 10 col strips (640)
  const int TN_N = NPAD / 64;                     // 18 col strips (1152)

  // 1) spatial mean -> x [NPAD,640] (rows >= 1100 zeroed)
  {
    int waves = NPAD * DMODEL;
    mean_kernel<<<(waves + WPB - 1) / WPB, blk, 0, stream>>>(support, query, x);
  }
  // 2) emb_q = x @ Wq^T, emb_v = x @ Wv^T   (zero rows propagate)
  {
    int strips = TM2 * TN_D;
    wmma_gemm_nt8<<<(strips + WPB - 1) / WPB, blk, 0, stream>>>(
        eq, x, Wq, TM2, TN_D, DMODEL, DMODEL, DMODEL, DMODEL, NPAD);
    wmma_gemm_nt8<<<(strips + WPB - 1) / WPB, blk, 0, stream>>>(
        ev, x, Wv, TM2, TN_D, DMODEL, DMODEL, DMODEL, DMODEL, NPAD);
  }
  // 3) Gram matrix G = emb_q @ emb_q^T (into w, NPAD x NPAD, padded part = 0)
  {
    int strips = TM2 * TN_N;
    wmma_gemm_nt8<<<(strips + WPB - 1) / WPB, blk, 0, stream>>>(
        w, eq, eq, TM2, TN_N, DMODEL, DMODEL, DMODEL, NPAD, NPAD);
  }
  // 4) squared norms
  rowsq_kernel<<<(NTOT + WPB - 1) / WPB, blk, 0, stream>>>(eq, sq);
  // 5) pairwise distances (in place over G)
  build_w_kernel<<<(NNP + 255) / 256, blk, 0, stream>>>(w, sq);
  // 6) mean / var (deterministic fixed-order reductions)
  red_sum_kernel<<<RED_BLOCKS, blk, 0, stream>>>(w, part);
  fin_mean_kernel<<<1, 1, 0, stream>>>(part, stats);
  red_var_kernel<<<RED_BLOCKS, blk, 0, stream>>>(w, stats, part);
  fin_var_kernel<<<1, 1, 0, stream>>>(part, stats);
  // 7) gaussian kernel (padded region forced back to 0)
  gauss_kernel<<<(NNP + 255) / 256, blk, 0, stream>>>(w, stats);
  // 8) top-k symmetric mask
  zero_mask_kernel<<<(NNP / 4 + 255) / 256, blk, 0, stream>>>(
      (unsigned int*)mask, NNP / 4);
  topk_mask_kernel<<<(NTOT + 255) / 256, blk, 0, stream>>>(w, mask);
  apply_mask_kernel<<<(NNP + 255) / 256, blk, 0, stream>>>(w, mask);
  // 9) degree + D^{-1/2}
  deg_kernel<<<(NTOT + WPB - 1) / WPB, blk, 0, stream>>>(w, deg);
  dsi_kernel<<<(NTOT + 255) / 256, blk, 0, stream>>>(deg, dsi);
  // 10) S = dsi * w * dsi, out = S @ emb_v (store only rows < 1100)
  scale_w_kernel<<<(NNP + 255) / 256, blk, 0, stream>>>(w, dsi);
  {
    int strips = TM2 * TN_D;
    wmma_gemm_nn8<<<(strips + WPB - 1) / WPB, blk, 0, stream>>>(
        out, w, ev, TM2, TN_D, NPAD, NPAD, DMODEL, DMODEL, NTOT);
  }
  (void)in_sizes; (void)n_in; (void)out_size; (void)ws_size;
}